// HSSM2D_58136677319325
// MI455X (gfx1250) — compile-verified
//
#include <hip/hip_runtime.h>
#include <math.h>

// Problem constants (match reference)
#define B_   16
#define C_   384
#define H_   64
#define W_   64
#define S_   16
#define GR_  64
#define HW_  (H_*W_)            // 4096
#define NPIX (B_*H_*W_)         // 65536
#define EPSF 1e-6f

#define PTILE 64                // pixels per block = one full (b,h) row
#define XST   72                // scalar LDS pixel stride (h1/hs tiles)
#define XP2   80                // pair-row stride (float2 units) for xs2: XP2*2 % 64 == 32
#define NPAIR (C_/2)            // 192 channel pairs

typedef float v2f __attribute__((ext_vector_type(2)));
typedef float v4f __attribute__((ext_vector_type(4)));
typedef float v8f __attribute__((ext_vector_type(8)));

// ---------------------------------------------------------------------------
// CDNA5 f32 WMMA: D(16x16) = A(16x4) * B(4x16) + C, fp32 throughout.
// A layout: vgpr j = {lanes0-15: K=j, lanes16-31: K=j+2}, M = lane%16.
// B layout: vgpr j = {lanes0-15: K=j, lanes16-31: K=j+2}, N = lane%16.
// C/D layout: vgpr v = {lanes0-15: M=v, lanes16-31: M=v+8}, N = lane%16.
// ---------------------------------------------------------------------------
__device__ __forceinline__ v8f wmma4(v2f a, v2f b, v8f c) {
  return __builtin_amdgcn_wmma_f32_16x16x4_f32(
      false, a, false, b, (short)0, c, false, false);
}

__device__ __forceinline__ float sigf(float x) { return 1.0f / (1.0f + __expf(-x)); }

// A-fragment pair load from channel-pair-interleaved LDS: one ds_load_b64.
__device__ __forceinline__ v2f ldsA(const float* xs2, int pairidx, int p) {
  return *(const v2f*)(xs2 + ((size_t)pairidx * XP2 + p) * 2);
}

// ---------------------------------------------------------------------------
// K0a: A matrices (Acol[s][t] = A[t][s]) and combined output bias.
// ---------------------------------------------------------------------------
__global__ void prep_kernel(float* __restrict__ acol, float* __restrict__ biasc,
                            const float* __restrict__ ral, const float* __restrict__ rU,
                            const float* __restrict__ rV,
                            const float* __restrict__ cal, const float* __restrict__ cU,
                            const float* __restrict__ cV,
                            const float* __restrict__ rbo, const float* __restrict__ rbs,
                            const float* __restrict__ cbo, const float* __restrict__ cbs) {
  int tid = threadIdx.x;
  if (tid < 256) {
    int t = tid & 15, s = tid >> 4;
    for (int d = 0; d < 2; ++d) {
      const float* al = d ? cal : ral;
      const float* U  = d ? cU  : rU;
      const float* V  = d ? cV  : rV;
      // A = diag(sigmoid(a)) + (U @ V^T)^T  ->  A[t][s] = d_ts*sig(a_t) + sum_r U[s,r]*V[t,r]
      float v = (t == s) ? sigf(al[t]) : 0.0f;
      for (int r = 0; r < 8; ++r) v += U[s*8 + r] * V[t*8 + r];
      acol[d*256 + s*16 + t] = v;
    }
  }
  if (tid < 384) biasc[tid] = 2.0f * (rbo[tid] + rbs[tid] + cbo[tid] + cbs[tid]);
}

// ---------------------------------------------------------------------------
// K0b: repack W (O x K row-major) into WMMA B-fragment order:
// dst[((chunk*Ksteps + kstep)*32 + lane)*2 + j]
//   = scale*(W1 + W2)[o = chunk*16 + lane%16][k = kstep*4 + j + (lane<16?0:2)]
// ---------------------------------------------------------------------------
__global__ void pack_kernel(float* __restrict__ dst, const float* __restrict__ s1,
                            const float* __restrict__ s2, int O, int K, float scale) {
  int ksteps = K >> 2;
  int total  = (O >> 4) * ksteps * 64;
  for (int idx = blockIdx.x * blockDim.x + threadIdx.x; idx < total;
       idx += gridDim.x * blockDim.x) {
    int chunk = idx / (ksteps * 64);
    int r     = idx % (ksteps * 64);
    int kst   = r >> 6;
    int r2    = r & 63;
    int lane  = r2 >> 1;
    int j     = r2 & 1;
    int o = chunk * 16 + (lane & 15);
    int k = kst * 4 + j + ((lane < 16) ? 0 : 2);
    float v = s1[o * K + k];
    if (s2) v += s2[o * K + k];
    dst[idx] = v * scale;
  }
}

// ---------------------------------------------------------------------------
// Stage one (b,h) row of x into channel-pair-interleaved LDS:
// xs2[cp][p] = {x[2cp][p], x[2cp+1][p]}, pair-row stride XP2 float2.
// ---------------------------------------------------------------------------
__device__ __forceinline__ void stage_x(float* xs2, const float* xb, int tid, int nthr) {
  for (int i = tid; i < NPAIR * 16; i += nthr) {
    int cp = i >> 4, seg = i & 15;
    const float* g0 = xb + (size_t)(2 * cp) * HW_ + seg * 4;
    v4f a  = *(const v4f*)g0;
    v4f b2 = *(const v4f*)(g0 + HW_);
    v4f lo = {a.x, b2.x, a.y, b2.y};
    v4f hi = {a.z, b2.z, a.w, b2.w};
    float* d = xs2 + ((size_t)cp * XP2 + seg * 4) * 2;
    *(v4f*)d       = lo;
    *(v4f*)(d + 4) = hi;
  }
}

// ---------------------------------------------------------------------------
// K1: GU = sigmoid(x@Wg^T + bg) * (x@Win^T + bin) for row and col params.
// 64-pixel tile per 128-thread block; each wave: one K=384 WMMA GEMM over
// 4 pixel sub-tiles (B fragment loaded once, used 4x).
// ---------------------------------------------------------------------------
__global__ __launch_bounds__(128) void gu_kernel(
    const float* __restrict__ x,
    const float* __restrict__ pwInR, const float* __restrict__ pwGR,
    const float* __restrict__ pwInC, const float* __restrict__ pwGC,
    const float* __restrict__ binR, const float* __restrict__ bgR,
    const float* __restrict__ binC, const float* __restrict__ bgC,
    float* __restrict__ GUr, float* __restrict__ GUc) {
  __shared__ float smem[NPAIR * XP2 * 2 + 4 * S_ * XST];   // xs2 | dump (~138 KB)
  float* xs2  = smem;
  float* dump = smem + NPAIR * XP2 * 2;                    // dump[g][s][p]

  int tid = threadIdx.x, lane = tid & 31, wave = tid >> 5;
  int P0 = blockIdx.x * PTILE;
  int b = P0 >> 12, h = (P0 & 4095) >> 6;
  const float* xb = x + (size_t)b * C_ * HW_ + h * W_;

  stage_x(xs2, xb, tid, 128);
  __syncthreads();

  const float* pw = (wave == 0) ? pwInR : (wave == 1) ? pwGR : (wave == 2) ? pwInC : pwGC;
  int hsel = (lane < 16) ? 0 : 1;
  int pl   = lane & 15;
  v8f acc[4] = {};
  for (int ks = 0; ks < 96; ++ks) {
    v2f bf = *(const v2f*)(pw + ((size_t)ks * 32 + lane) * 2);
    int pairidx = ks * 2 + hsel;
#pragma unroll
    for (int sub = 0; sub < 4; ++sub)
      acc[sub] = wmma4(ldsA(xs2, pairidx, sub * 16 + pl), bf, acc[sub]);
  }
  {
    int s = lane & 15;
    int pr = (lane < 16) ? 0 : 8;
#pragma unroll
    for (int sub = 0; sub < 4; ++sub)
      for (int v = 0; v < 8; ++v)
        dump[wave * S_ * XST + s * XST + sub * 16 + pr + v] = acc[sub][v];
  }
  __syncthreads();

  for (int e = tid; e < PTILE * S_; e += 128) {
    int s = e & 15, pp = e >> 4;
    float ur = dump[0 * S_ * XST + s * XST + pp] + binR[s];
    float gr = sigf(dump[1 * S_ * XST + s * XST + pp] + bgR[s]);
    GUr[(size_t)P0 * 16 + e] = gr * ur;              // row-order = flat pixel order
    float uc = dump[2 * S_ * XST + s * XST + pp] + binC[s];
    float gc = sigf(dump[3 * S_ * XST + s * XST + pp] + bgC[s]);
    int q = b * HW_ + pp * 64 + h;                   // col-order pixel index
    GUc[(size_t)q * 16 + s] = gc * uc;
  }
}

// ---------------------------------------------------------------------------
// K2: bidirectional 16-state scan. HS[l] = h_fwd[l] + h_rev[L-1-l].
// 4 sequences / 64-thread block; 16 lanes per sequence; A column in regs;
// cross-lane h via wave shuffles (wave-synchronous, no barriers in the loop).
// ---------------------------------------------------------------------------
#define GU_STRIDE 1040  // 64*16 + 16 pad
__global__ __launch_bounds__(64) void scan_kernel(
    const float* __restrict__ GUr, const float* __restrict__ GUc,
    float* __restrict__ HSr, float* __restrict__ HSc,
    const float* __restrict__ acol) {
  __shared__ float gu[4 * GU_STRIDE];
  __shared__ float hsb[4 * GU_STRIDE];
  int tid = threadIdx.x;
  int dir = blockIdx.y;
  const float* GU = dir ? GUc : GUr;
  float* HS = dir ? HSc : HSr;
  int seqbase = blockIdx.x * 4;

  float areg[16];
  {
    int s = tid & 15;
    for (int t = 0; t < 16; ++t) areg[t] = acol[dir * 256 + s * 16 + t];
  }
  const float* src = GU + (size_t)seqbase * 1024;
  for (int it = 0; it < 16; ++it) {
    int fi = tid + it * 64;
    int seq = fi >> 8, r4 = fi & 255;
    *(v4f*)(gu + seq * GU_STRIDE + r4 * 4) = *(const v4f*)(src + (size_t)fi * 4);
  }
  __syncthreads();

  int seq = tid >> 4;
  int s   = tid & 15;
  int hb  = tid & 16;

  float h = 0.f;
  for (int j = 0; j < 64; ++j) {        // reverse pass
    int l = 63 - j;
    float hn = gu[seq * GU_STRIDE + l * 16 + s];
    for (int t = 0; t < 16; ++t) hn += __shfl(h, hb + t, 32) * areg[t];
    h = hn;
    hsb[seq * GU_STRIDE + l * 16 + s] = h;
  }
  h = 0.f;
  for (int l = 0; l < 64; ++l) {        // forward pass + combine + store
    float hn = gu[seq * GU_STRIDE + l * 16 + s];
    for (int t = 0; t < 16; ++t) hn += __shfl(h, hb + t, 32) * areg[t];
    h = hn;
    HS[((size_t)(seqbase + seq) * 64 + l) * 16 + s] = h + hsb[seq * GU_STRIDE + l * 16 + s];
  }
}

// ---------------------------------------------------------------------------
// K3: mega-fused output, 64-pixel tile per 256-thread block.
//  y   = 0.25*( HSr@WoutR^T + HSc@WoutC^T + x@Wsum^T + biasc )
//  xn  = layernorm_C(x)*nw + nb ; h1 = gelu(xn@g1^T + b1)
//  g   = sigmoid(h1@g2^T + b2)  ; out = g*y + (1-g)*x
// ---------------------------------------------------------------------------
#define L_XS   0                         // xs2: NPAIR*XP2*2 floats
#define L_H1   (NPAIR * XP2 * 2)         // h1s: GR_*XST (reused for LN partials)
#define L_HSR  (L_H1 + GR_ * XST)        // hsr: S_*XST
#define L_HSC  (L_HSR + S_ * XST)        // hsc: S_*XST
#define L_NW   (L_HSC + S_ * XST)        // nw: C_
#define L_NB   (L_NW + C_)               // nb: C_
#define L_MEAN (L_NB + C_)               // meanv: 64
#define L_RSTD (L_MEAN + 64)             // rstdv: 64
#define L_TOT  (L_RSTD + 64)             // ~150 KB (gfx1250: 320 KB/WGP)

__global__ __launch_bounds__(256) void fused_kernel(
    const float* __restrict__ x,
    const float* __restrict__ HSr, const float* __restrict__ HSc,
    const float* __restrict__ pwSum, const float* __restrict__ pwOutR,
    const float* __restrict__ pwOutC, const float* __restrict__ pwG1,
    const float* __restrict__ pwG2, const float* __restrict__ biasc,
    const float* __restrict__ g1b, const float* __restrict__ g2b,
    const float* __restrict__ nwg, const float* __restrict__ nbg,
    float* __restrict__ out) {
  __shared__ float smem[L_TOT];
  float* xs2   = smem + L_XS;
  float* h1s   = smem + L_H1;
  float* hsr   = smem + L_HSR;
  float* hsc   = smem + L_HSC;
  float* nw    = smem + L_NW;
  float* nb    = smem + L_NB;
  float* meanv = smem + L_MEAN;
  float* rstdv = smem + L_RSTD;
  float* psum  = h1s;                  // reuse before h1 is produced
  float* psq   = h1s + 256;

  int tid = threadIdx.x, lane = tid & 31, wave = tid >> 5;
  int P0 = blockIdx.x * PTILE;
  int b = P0 >> 12, h = (P0 & 4095) >> 6;
  const float* xb = x + (size_t)b * C_ * HW_ + h * W_;

  stage_x(xs2, xb, tid, 256);
  for (int i = tid; i < C_; i += 256) { nw[i] = nwg[i]; nb[i] = nbg[i]; }
  for (int e = tid; e < PTILE * S_; e += 256) {
    int s = e & 15, pp = e >> 4;
    hsr[s * XST + pp] = HSr[(size_t)P0 * 16 + e];
    int q = b * HW_ + pp * 64 + h;
    hsc[s * XST + pp] = HSc[(size_t)q * 16 + s];
  }
  __syncthreads();

  {  // per-pixel layernorm stats (4 partials x 64 pixels, 48 channel pairs each)
    int part = tid >> 6, pp = tid & 63;
    float sm = 0.f, sq = 0.f;
    for (int cp = part * 48; cp < part * 48 + 48; ++cp) {
      v2f v = ldsA(xs2, cp, pp);
      sm += v.x + v.y; sq += v.x * v.x + v.y * v.y;
    }
    psum[part * 64 + pp] = sm; psq[part * 64 + pp] = sq;
  }
  __syncthreads();
  if (tid < 64) {
    float sm = 0.f, sq = 0.f;
    for (int part = 0; part < 4; ++part) { sm += psum[part * 64 + tid]; sq += psq[part * 64 + tid]; }
    float m = sm / 384.f;
    meanv[tid] = m;
    rstdv[tid] = rsqrtf(sq / 384.f - m * m + EPSF);
  }
  __syncthreads();

  int hsel = (lane < 16) ? 0 : 1;
  int off  = hsel * 2;
  int pl   = lane & 15;

  // GEMM2: h1 (chunk = wave<4), xn formed on the fly, 4 pixel sub-tiles
  v8f hacc[4] = {};
  if (wave < 4) {
    float mp[4], rp[4];
#pragma unroll
    for (int sub = 0; sub < 4; ++sub) {
      mp[sub] = meanv[sub * 16 + pl];
      rp[sub] = rstdv[sub * 16 + pl];
    }
    for (int ks = 0; ks < 96; ++ks) {
      int c0 = ks * 4 + off;
      float w0 = nw[c0], w1 = nw[c0 + 1], b0 = nb[c0], b1 = nb[c0 + 1];
      v2f bf = *(const v2f*)(pwG1 + ((size_t)(wave * 96 + ks) * 32 + lane) * 2);
      int pairidx = ks * 2 + hsel;
#pragma unroll
      for (int sub = 0; sub < 4; ++sub) {
        v2f xv = ldsA(xs2, pairidx, sub * 16 + pl);
        v2f a;
        a.x = (xv.x - mp[sub]) * rp[sub] * w0 + b0;
        a.y = (xv.y - mp[sub]) * rp[sub] * w1 + b1;
        hacc[sub] = wmma4(a, bf, hacc[sub]);
      }
    }
  }
  __syncthreads();   // LN partials no longer needed; safe to overwrite h1s
  if (wave < 4) {
    int o = wave * 16 + (lane & 15);
    float bb = g1b[o];
    int pr = (lane < 16) ? 0 : 8;
#pragma unroll
    for (int sub = 0; sub < 4; ++sub)
      for (int v = 0; v < 8; ++v) {
        float z = hacc[sub][v] + bb;
        h1s[o * XST + sub * 16 + pr + v] = 0.5f * z * (1.0f + erff(z * 0.70710678118f));
      }
  }
  __syncthreads();

  for (int ci = 0; ci < 3; ++ci) {  // 24 output chunks / 8 waves
    int chunk = wave * 3 + ci;
    v8f yacc[4] = {};
    for (int ks = 0; ks < 96; ++ks) {          // x @ Wsum^T (K=384)
      v2f bf = *(const v2f*)(pwSum + ((size_t)(chunk * 96 + ks) * 32 + lane) * 2);
      int pairidx = ks * 2 + hsel;
#pragma unroll
      for (int sub = 0; sub < 4; ++sub)
        yacc[sub] = wmma4(ldsA(xs2, pairidx, sub * 16 + pl), bf, yacc[sub]);
    }
    for (int ks = 0; ks < 4; ++ks) {           // HSr @ WoutR^T (K=16)
      int c0 = ks * 4 + off;
      v2f bf = *(const v2f*)(pwOutR + ((size_t)(chunk * 4 + ks) * 32 + lane) * 2);
#pragma unroll
      for (int sub = 0; sub < 4; ++sub) {
        v2f a;
        a.x = hsr[c0 * XST + sub * 16 + pl];
        a.y = hsr[(c0 + 1) * XST + sub * 16 + pl];
        yacc[sub] = wmma4(a, bf, yacc[sub]);
      }
    }
    for (int ks = 0; ks < 4; ++ks) {           // HSc @ WoutC^T (K=16)
      int c0 = ks * 4 + off;
      v2f bf = *(const v2f*)(pwOutC + ((size_t)(chunk * 4 + ks) * 32 + lane) * 2);
#pragma unroll
      for (int sub = 0; sub < 4; ++sub) {
        v2f a;
        a.x = hsc[c0 * XST + sub * 16 + pl];
        a.y = hsc[(c0 + 1) * XST + sub * 16 + pl];
        yacc[sub] = wmma4(a, bf, yacc[sub]);
      }
    }
    v8f gacc[4] = {};
    for (int ks = 0; ks < 16; ++ks) {          // h1 @ g2^T (K=64)
      int c0 = ks * 4 + off;
      v2f bf = *(const v2f*)(pwG2 + ((size_t)(chunk * 16 + ks) * 32 + lane) * 2);
#pragma unroll
      for (int sub = 0; sub < 4; ++sub) {
        v2f a;
        a.x = h1s[c0 * XST + sub * 16 + pl];
        a.y = h1s[(c0 + 1) * XST + sub * 16 + pl];
        gacc[sub] = wmma4(a, bf, gacc[sub]);
      }
    }
    int o = chunk * 16 + (lane & 15);
    float bc = biasc[o], b2 = g2b[o];
    int pr = (lane < 16) ? 0 : 8;
    const float* xrow = xs2 + (size_t)(o >> 1) * XP2 * 2 + (o & 1);
    float* op = out + (size_t)b * C_ * HW_ + (size_t)o * HW_ + h * W_ + pr;
#pragma unroll
    for (int sub = 0; sub < 4; ++sub) {
      float res[8];
      for (int v = 0; v < 8; ++v) {
        float y  = 0.25f * (yacc[sub][v] + bc);
        float g  = sigf(gacc[sub][v] + b2);
        float xv = xrow[(sub * 16 + pr + v) * 2];
        res[v] = g * y + (1.0f - g) * xv;
      }
      v4f lo = {res[0], res[1], res[2], res[3]};
      v4f hi = {res[4], res[5], res[6], res[7]};
      *(v4f*)(op + sub * 16)     = lo;
      *(v4f*)(op + sub * 16 + 4) = hi;
    }
  }
}

// ---------------------------------------------------------------------------
// Host launcher. Workspace layout (floats), total ~17.7 MB.
// ---------------------------------------------------------------------------
extern "C" void kernel_launch(void* const* d_in, const int* in_sizes, int n_in,
                              void* d_out, int out_size, void* d_ws, size_t ws_size,
                              hipStream_t stream) {
  const float* x       = (const float*)d_in[0];
  const float* r_alog  = (const float*)d_in[1];
  const float* r_U     = (const float*)d_in[2];
  const float* r_V     = (const float*)d_in[3];
  const float* r_Win   = (const float*)d_in[4];
  const float* r_bin   = (const float*)d_in[5];
  const float* r_Wg    = (const float*)d_in[6];
  const float* r_bg    = (const float*)d_in[7];
  const float* r_Wout  = (const float*)d_in[8];
  const float* r_bout  = (const float*)d_in[9];
  const float* r_Wskip = (const float*)d_in[10];
  const float* r_bskip = (const float*)d_in[11];
  const float* c_alog  = (const float*)d_in[12];
  const float* c_U     = (const float*)d_in[13];
  const float* c_V     = (const float*)d_in[14];
  const float* c_Win   = (const float*)d_in[15];
  const float* c_bin   = (const float*)d_in[16];
  const float* c_Wg    = (const float*)d_in[17];
  const float* c_bg    = (const float*)d_in[18];
  const float* c_Wout  = (const float*)d_in[19];
  const float* c_bout  = (const float*)d_in[20];
  const float* c_Wskip = (const float*)d_in[21];
  const float* c_bskip = (const float*)d_in[22];
  const float* g1_w    = (const float*)d_in[23];
  const float* g1_b    = (const float*)d_in[24];
  const float* g2_w    = (const float*)d_in[25];
  const float* g2_b    = (const float*)d_in[26];
  const float* norm_w  = (const float*)d_in[27];
  const float* norm_b  = (const float*)d_in[28];
  float* out = (float*)d_out;

  float* ws = (float*)d_ws;
  float* Acol   = ws;                 // 512
  float* biasc  = Acol + 512;         // 384
  float* pwSum  = biasc + 384;        // 147456
  float* pwOutR = pwSum + 147456;     // 6144
  float* pwOutC = pwOutR + 6144;      // 6144
  float* pwG1   = pwOutC + 6144;      // 24576
  float* pwG2   = pwG1 + 24576;       // 24576
  float* pwInR  = pwG2 + 24576;       // 6144
  float* pwGR   = pwInR + 6144;       // 6144
  float* pwInC  = pwGR + 6144;        // 6144
  float* pwGC   = pwInC + 6144;       // 6144
  float* GUr    = pwGC + 6144;        // 1048576
  float* GUc    = GUr + (size_t)NPIX * 16;
  float* HSr    = GUc + (size_t)NPIX * 16;
  float* HSc    = HSr + (size_t)NPIX * 16;

  // K0: prep A / bias / packed weights
  prep_kernel<<<1, 384, 0, stream>>>(Acol, biasc, r_alog, r_U, r_V, c_alog, c_U, c_V,
                                     r_bout, r_bskip, c_bout, c_bskip);
  pack_kernel<<<576, 256, 0, stream>>>(pwSum, r_Wskip, c_Wskip, 384, 384, 2.0f);
  pack_kernel<<<24, 256, 0, stream>>>(pwOutR, r_Wout, nullptr, 384, 16, 1.0f);
  pack_kernel<<<24, 256, 0, stream>>>(pwOutC, c_Wout, nullptr, 384, 16, 1.0f);
  pack_kernel<<<96, 256, 0, stream>>>(pwG1, g1_w, nullptr, 64, 384, 1.0f);
  pack_kernel<<<96, 256, 0, stream>>>(pwG2, g2_w, nullptr, 384, 64, 1.0f);
  pack_kernel<<<24, 256, 0, stream>>>(pwInR, r_Win, nullptr, 16, 384, 1.0f);
  pack_kernel<<<24, 256, 0, stream>>>(pwGR, r_Wg, nullptr, 16, 384, 1.0f);
  pack_kernel<<<24, 256, 0, stream>>>(pwInC, c_Win, nullptr, 16, 384, 1.0f);
  pack_kernel<<<24, 256, 0, stream>>>(pwGC, c_Wg, nullptr, 16, 384, 1.0f);

  // K1: GU for row and col parameter sets (reads x once)
  gu_kernel<<<NPIX / PTILE, 128, 0, stream>>>(x, pwInR, pwGR, pwInC, pwGC,
                                              r_bin, r_bg, c_bin, c_bg, GUr, GUc);

  // K2: bidirectional scans (row: y=0, col: y=1)
  scan_kernel<<<dim3(256, 2), 64, 0, stream>>>(GUr, GUc, HSr, HSc, Acol);

  // K3: fused output (reads x once more, writes out once)
  fused_kernel<<<NPIX / PTILE, 256, 0, stream>>>(x, HSr, HSc, pwSum, pwOutR, pwOutC,
                                                 pwG1, pwG2, biasc, g1_b, g2_b,
                                                 norm_w, norm_b, out);
}